// SimpleGCN_19241453486766
// MI455X (gfx1250) — compile-verified
//
#include <hip/hip_runtime.h>

typedef __attribute__((ext_vector_type(2))) float v2f;
typedef __attribute__((ext_vector_type(8))) float v8f;

#define D 64
#define D_OUT 10
#define N_GRAPHS 256
#define LDSP 68   // padded LDS row pitch (floats): bank-conflict-free b64 reads

// ---------------------------------------------------------------------------
// Deterministic hash RNG for dropout (JAX threefry cannot be bit-matched;
// mask is deterministic per call as required by the harness contract).
// ---------------------------------------------------------------------------
__device__ __forceinline__ float hash_uniform(unsigned x) {
  x ^= x >> 16; x *= 0x7feb352du;
  x ^= x >> 15; x *= 0x846ca68bu;
  x ^= x >> 16;
  return (float)(x >> 8) * (1.0f / 16777216.0f);
}

__device__ __forceinline__ float dropout_keep(float v, unsigned idx, unsigned seed) {
  float u = hash_uniform(idx * 0x9e3779b9u + seed);
  return (u < 0.7f) ? v * (1.0f / 0.7f) : 0.0f;   // rate=0.3 -> keep=0.7
}

// ---------------------------------------------------------------------------
// Degree / symmetric-norm kernels
// ---------------------------------------------------------------------------
__global__ void deg_init(float* __restrict__ dinv, int n) {
  int i = blockIdx.x * blockDim.x + threadIdx.x;
  if (i < n) dinv[i] = 1.0f;                       // self-loop contribution
}

__global__ void deg_count(const long long* __restrict__ dst, float* __restrict__ dinv, int E) {
  int e = blockIdx.x * blockDim.x + threadIdx.x;
  if (e < E) atomicAdd(&dinv[(int)dst[e]], 1.0f);
}

__global__ void deg_finish(float* __restrict__ dinv, int n) {
  int i = blockIdx.x * blockDim.x + threadIdx.x;
  if (i < n) dinv[i] = rsqrtf(dinv[i]);
}

// ---------------------------------------------------------------------------
// H = X @ W  (n x 64) @ (64 x 64), full f32 via V_WMMA_F32_16X16X4_F32.
// One wave -> one 16-row tile x all 64 cols (4 column tiles, 4 accumulators).
// Block = 256 threads = 8 waves = 128 rows.
// W is staged TRANSPOSED in LDS (sWt[col][k], pitch 68) so each B fragment
// (W[k][col], W[k+1][col]) is one contiguous, conflict-free ds_load_b64.
// ---------------------------------------------------------------------------
__global__ __launch_bounds__(256) void gemm_n64(const float* __restrict__ X,
                                                const float* __restrict__ W,
                                                float* __restrict__ H, int n) {
  __shared__ float sWt[D * LDSP];
  int tid = threadIdx.x;
  #pragma unroll
  for (int i = tid * 4; i < D * D; i += 256 * 4) {
    float4 w = *(const float4*)(&W[i]);            // coalesced row-major read
    int k = i >> 6, c = i & 63;
    sWt[(c + 0) * LDSP + k] = w.x;                 // transposed store (once)
    sWt[(c + 1) * LDSP + k] = w.y;
    sWt[(c + 2) * LDSP + k] = w.z;
    sWt[(c + 3) * LDSP + k] = w.w;
  }
  __syncthreads();

  int wave  = tid >> 5;
  int lane  = tid & 31;
  int m     = lane & 15;     // A: row-in-tile; B/C/D: col-in-tile
  int khalf = lane >> 4;     // which K-pair of the 16x16x4 fragment
  int tile_base = blockIdx.x * 128 + wave * 16;

  int arow  = tile_base + m;
  int arowc = arow < n ? arow : (n - 1);           // clamp, keep EXEC all-ones
  const float* xrow = X + (size_t)arowc * D;
  const float* bcol = &sWt[m * LDSP + 2 * khalf];  // lane's B column base

  v8f acc0 = {}, acc1 = {}, acc2 = {}, acc3 = {};
  #pragma unroll
  for (int k0 = 0; k0 < D; k0 += 4) {
    v2f a = *(const v2f*)(&xrow[k0 + 2 * khalf]);  // global_load_b64
    v2f b0 = *(const v2f*)(&bcol[k0]);             // ds_load_b64, imm offsets
    v2f b1 = *(const v2f*)(&bcol[16 * LDSP + k0]);
    v2f b2 = *(const v2f*)(&bcol[32 * LDSP + k0]);
    v2f b3 = *(const v2f*)(&bcol[48 * LDSP + k0]);
    acc0 = __builtin_amdgcn_wmma_f32_16x16x4_f32(false, a, false, b0, (short)0, acc0, false, false);
    acc1 = __builtin_amdgcn_wmma_f32_16x16x4_f32(false, a, false, b1, (short)0, acc1, false, false);
    acc2 = __builtin_amdgcn_wmma_f32_16x16x4_f32(false, a, false, b2, (short)0, acc2, false, false);
    acc3 = __builtin_amdgcn_wmma_f32_16x16x4_f32(false, a, false, b3, (short)0, acc3, false, false);
  }

  // C/D layout: VGPR v holds row M = v + 8*khalf, col N = lane&15
  #pragma unroll
  for (int v = 0; v < 8; ++v) {
    int orow = tile_base + v + 8 * khalf;
    if (orow < n) {
      float* hp = H + (size_t)orow * D;
      hp[m]      = acc0[v];
      hp[16 + m] = acc1[v];
      hp[32 + m] = acc2[v];
      hp[48 + m] = acc3[v];
    }
  }
}

// ---------------------------------------------------------------------------
// AGG = H * dinv^2   (self-loop term folded into the aggregation init)
// ---------------------------------------------------------------------------
__global__ void selfloop_init(const float* __restrict__ H, const float* __restrict__ dinv,
                              float* __restrict__ AGG, int n) {
  int idx = blockIdx.x * blockDim.x + threadIdx.x;
  if (idx < n * D) {
    float di = dinv[idx >> 6];
    AGG[idx] = H[idx] * di * di;
  }
}

// ---------------------------------------------------------------------------
// Edge scatter: AGG[dst] += H[src] * dinv[src]*dinv[dst].
// 16 lanes per edge, each lane: one float4 gather + 4 f32 atomics (L2-resident).
// ---------------------------------------------------------------------------
__global__ __launch_bounds__(256) void scatter_edges(const float* __restrict__ H,
                                                     const long long* __restrict__ src,
                                                     const long long* __restrict__ dst,
                                                     const float* __restrict__ dinv,
                                                     float* __restrict__ AGG, int E) {
  long long idx = (long long)blockIdx.x * blockDim.x + threadIdx.x;
  int e = (int)(idx >> 4);
  if (e >= E) return;
  int c = ((int)idx & 15) * 4;
  int s = (int)src[e];
  int d = (int)dst[e];
  float norm = dinv[s] * dinv[d];
  float4 v = *(const float4*)(H + (size_t)s * D + c);
  float* ap = AGG + (size_t)d * D + c;
  atomicAdd(ap + 0, v.x * norm);
  atomicAdd(ap + 1, v.y * norm);
  atomicAdd(ap + 2, v.z * norm);
  atomicAdd(ap + 3, v.w * norm);
}

// ---------------------------------------------------------------------------
// OUT = act(AGG + b);  act = ReLU + dropout for hidden layers, identity last.
// ---------------------------------------------------------------------------
__global__ void epilogue(const float* __restrict__ AGG, const float* __restrict__ bias,
                         float* __restrict__ OUT, int n, int relu_drop, unsigned seed) {
  int idx = blockIdx.x * blockDim.x + threadIdx.x;
  if (idx >= n * D) return;
  float v = AGG[idx] + bias[idx & 63];
  if (relu_drop) {
    v = fmaxf(v, 0.0f);
    v = dropout_keep(v, (unsigned)idx, seed);
  }
  OUT[idx] = v;
}

// ---------------------------------------------------------------------------
// Global mean pool + dropout + final linear
// ---------------------------------------------------------------------------
__global__ void pool_zero(float* __restrict__ sums, float* __restrict__ cnt) {
  int i = blockIdx.x * blockDim.x + threadIdx.x;
  if (i < N_GRAPHS * D) sums[i] = 0.0f;
  if (i < N_GRAPHS) cnt[i] = 0.0f;
}

__global__ void pool_accum(const float* __restrict__ H, const long long* __restrict__ batch,
                           float* __restrict__ sums, float* __restrict__ cnt, int n) {
  int idx = blockIdx.x * blockDim.x + threadIdx.x;
  if (idx >= n * D) return;
  int i = idx >> 6, f = idx & 63;
  int g = (int)batch[i];
  atomicAdd(&sums[g * D + f], H[idx]);
  if (f == 0) atomicAdd(&cnt[g], 1.0f);
}

__global__ void pool_final(const float* __restrict__ sums, const float* __restrict__ cnt,
                           float* __restrict__ pooled, unsigned seed) {
  int idx = blockIdx.x * blockDim.x + threadIdx.x;
  if (idx >= N_GRAPHS * D) return;
  float v = sums[idx] / fmaxf(cnt[idx >> 6], 1.0f);
  pooled[idx] = dropout_keep(v, (unsigned)idx, seed);
}

__global__ void final_linear(const float* __restrict__ pooled, const float* __restrict__ Wl,
                             const float* __restrict__ bl, float* __restrict__ out) {
  int idx = blockIdx.x * blockDim.x + threadIdx.x;
  if (idx >= N_GRAPHS * D_OUT) return;
  int g = idx / D_OUT, o = idx - g * D_OUT;
  float acc = bl[o];
  #pragma unroll
  for (int f = 0; f < D; ++f) acc += pooled[g * D + f] * Wl[f * D_OUT + o];
  out[idx] = acc;
}

// ---------------------------------------------------------------------------
// Launch: deg -> 3x(gemm, init, scatter, epilogue) -> pool -> linear
// Workspace: dinv (256KB slot) | bufA (12.8MB) | bufB (12.8MB) | sums | cnt | pooled
// ---------------------------------------------------------------------------
extern "C" void kernel_launch(void* const* d_in, const int* in_sizes, int n_in,
                              void* d_out, int out_size, void* d_ws, size_t ws_size,
                              hipStream_t stream) {
  const float*     x     = (const float*)d_in[0];
  const long long* ei    = (const long long*)d_in[1];   // int64 [2, E]
  const long long* batch = (const long long*)d_in[2];   // int64 [N]
  const float* W1 = (const float*)d_in[3]; const float* b1 = (const float*)d_in[4];
  const float* W2 = (const float*)d_in[5]; const float* b2 = (const float*)d_in[6];
  const float* W3 = (const float*)d_in[7]; const float* b3 = (const float*)d_in[8];
  const float* Wl = (const float*)d_in[9]; const float* bl = (const float*)d_in[10];
  float* out = (float*)d_out;

  int E = in_sizes[1] / 2;
  int n = in_sizes[0] / D;
  const long long* src = ei;
  const long long* dstp = ei + E;

  float* ws     = (float*)d_ws;
  float* dinv   = ws;
  float* bufA   = ws + 65536;
  float* bufB   = bufA + (size_t)n * D;
  float* sums   = bufB + (size_t)n * D;
  float* cnt    = sums + N_GRAPHS * D;
  float* pooled = cnt + N_GRAPHS;

  dim3 blk(256);
  dim3 gN((n + 255) / 256);
  dim3 gNd((n * D + 255) / 256);
  dim3 gE((E + 255) / 256);
  dim3 gE16((unsigned)((16LL * E + 255) / 256));
  dim3 gG((N_GRAPHS * D + 255) / 256);
  dim3 gGo((N_GRAPHS * D_OUT + 255) / 256);
  dim3 gGemm((n + 127) / 128);

  // degree -> dinv = rsqrt(deg_dst + 1)
  deg_init  <<<gN,  blk, 0, stream>>>(dinv, n);
  deg_count <<<gE,  blk, 0, stream>>>(dstp, dinv, E);
  deg_finish<<<gN,  blk, 0, stream>>>(dinv, n);

  // layer 1
  gemm_n64     <<<gGemm, blk, 0, stream>>>(x, W1, bufA, n);
  selfloop_init<<<gNd,   blk, 0, stream>>>(bufA, dinv, bufB, n);
  scatter_edges<<<gE16,  blk, 0, stream>>>(bufA, src, dstp, dinv, bufB, E);
  epilogue     <<<gNd,   blk, 0, stream>>>(bufB, b1, bufA, n, 1, 0x11111111u);

  // layer 2
  gemm_n64     <<<gGemm, blk, 0, stream>>>(bufA, W2, bufB, n);
  selfloop_init<<<gNd,   blk, 0, stream>>>(bufB, dinv, bufA, n);
  scatter_edges<<<gE16,  blk, 0, stream>>>(bufB, src, dstp, dinv, bufA, E);
  epilogue     <<<gNd,   blk, 0, stream>>>(bufA, b2, bufB, n, 1, 0x22222222u);

  // layer 3 (no relu/dropout)
  gemm_n64     <<<gGemm, blk, 0, stream>>>(bufB, W3, bufA, n);
  selfloop_init<<<gNd,   blk, 0, stream>>>(bufA, dinv, bufB, n);
  scatter_edges<<<gE16,  blk, 0, stream>>>(bufA, src, dstp, dinv, bufB, E);
  epilogue     <<<gNd,   blk, 0, stream>>>(bufB, b3, bufA, n, 0, 0u);

  // pool + dropout + linear
  pool_zero   <<<gG,  blk, 0, stream>>>(sums, cnt);
  pool_accum  <<<gNd, blk, 0, stream>>>(bufA, batch, sums, cnt, n);
  pool_final  <<<gG,  blk, 0, stream>>>(sums, cnt, pooled, 0x33333333u);
  final_linear<<<gGo, blk, 0, stream>>>(pooled, Wl, bl, out);
}